// CoAttention_64639257805493
// MI455X (gfx1250) — compile-verified
//
#include <hip/hip_runtime.h>
#include <hip/hip_bf16.h>

// CoAttention for MI455X (gfx1250), wave32, WMMA f32 16x16x4 path.
// B=16, LC=512, LQ=64, D=256, H=256.

typedef __attribute__((ext_vector_type(2))) float v2f;
typedef __attribute__((ext_vector_type(8))) float v8f;
typedef int v4i __attribute__((vector_size(16)));   // matches builtin's V4i param

#define B_  16
#define LC_ 512
#define LQ_ 64
#define D_  256
#define H_  256

__device__ __forceinline__ float sigf(float v) { return 1.f / (1.f + __expf(-v)); }

// ---- gfx1250 async global->LDS copy (ASYNCcnt), with safe fallback --------
#if __has_builtin(__builtin_amdgcn_global_load_async_to_lds_b128) && \
    __has_builtin(__builtin_amdgcn_s_wait_asynccnt)
#define HAVE_ASYNC_LDS 1
#else
#define HAVE_ASYNC_LDS 0
#endif

__device__ __forceinline__ void cp16_g2l(float* lds, const float* g) {
#if HAVE_ASYNC_LDS
    __builtin_amdgcn_global_load_async_to_lds_b128(
        (__attribute__((address_space(1))) v4i*)g,
        (__attribute__((address_space(3))) v4i*)lds, 0, 0);
#else
    *(float4*)lds = *(const float4*)g;
#endif
}

__device__ __forceinline__ void cp_wait() {
#if HAVE_ASYNC_LDS
    __builtin_amdgcn_s_wait_asynccnt(0);
#endif
}

// ---------------------------------------------------------------------------
// fp32 GEMM:  Y[M,N] = X[M,K] @ (TRANSB ? W[N,K]^T : W[K,N]) + bias
// ACT==1: Y = GX * sigmoid(Y)   (gate epilogue; N == K layout of GX)
// Block = 256 thr = 8 waves; block tile 128(M) x 64(N); K staged in LDS in
// double-buffered 32-wide chunks via async global->LDS b128 copies.
// Wave computes 16x64 via 4 accumulators of v_wmma_f32_16x16x4_f32.
// grid = (N/64, M/128). Requires K % 32 == 0.
// ---------------------------------------------------------------------------
template<int ACT, int TRANSB>
__global__ __launch_bounds__(256) void gemm_kernel(
    const float* __restrict__ X, const float* __restrict__ W,
    const float* __restrict__ bias, const float* __restrict__ GX,
    float* __restrict__ Y, int N, int K)
{
    constexpr int KC = 32;
    // A: 128 rows x 32 k, row pad to 36 (16B-aligned vec4 stores, conflict-free reads)
    __shared__ float As[2][128 * 36];
    // B: TRANSB -> [nn][kk] padded to 36; else [kk][nn] (64 contiguous)
    __shared__ float Bs[2][TRANSB ? (64 * 36) : (KC * 64)];

    const int tid  = threadIdx.x;
    const int lane = tid & 31;
    const int wave = tid >> 5;
    const int ln   = lane & 15;
    const int kh   = (lane < 16) ? 0 : 2;        // K-half per WMMA A/B layout
    const int mB   = blockIdx.y * 128;           // block M base
    const int m0   = mB + wave * 16;             // wave M base
    const int n0   = blockIdx.x * 64;
    const int rA   = wave * 16 + ln;             // A row within block tile

    v8f acc[4] = {v8f{}, v8f{}, v8f{}, v8f{}};

    auto stage = [&](int k0, int bufw) {
        // A chunk: 128x32 floats, 4 x vec4 per thread, coalesced along k.
        #pragma unroll
        for (int i = 0; i < 4; ++i) {
            const int linear = i * 256 + tid;
            const int row = linear >> 3;
            const int kq  = (linear & 7) * 4;
            cp16_g2l(&As[bufw][row * 36 + kq],
                     X + (size_t)(mB + row) * K + k0 + kq);
        }
        // B chunk: 32x64 floats, 2 x vec4 per thread.
        #pragma unroll
        for (int i = 0; i < 2; ++i) {
            const int linear = i * 256 + tid;
            if (TRANSB) {
                const int nn = linear >> 3;
                const int kq = (linear & 7) * 4;
                cp16_g2l(&Bs[bufw][nn * 36 + kq],
                         W + (size_t)(n0 + nn) * K + k0 + kq);
            } else {
                const int kk = linear >> 4;
                const int nq = (linear & 15) * 4;
                cp16_g2l(&Bs[bufw][kk * 64 + nq],
                         W + (size_t)(k0 + kk) * N + n0 + nq);
            }
        }
    };

    const int nch = K / KC;
    stage(0, 0);
    cp_wait();
    __syncthreads();

    int buf = 0;
    for (int c = 0; c < nch; ++c) {
        if (c + 1 < nch) stage((c + 1) * KC, buf ^ 1);   // overlap copy w/ WMMA
        #pragma unroll
        for (int kk = 0; kk < KC; kk += 4) {
            v2f a;
            a.x = As[buf][rA * 36 + kk + kh];
            a.y = As[buf][rA * 36 + kk + kh + 1];
            #pragma unroll
            for (int t = 0; t < 4; ++t) {
                const int nn = t * 16 + ln;
                v2f bf;
                if (TRANSB) {
                    bf.x = Bs[buf][nn * 36 + kk + kh];
                    bf.y = Bs[buf][nn * 36 + kk + kh + 1];
                } else {
                    bf.x = Bs[buf][(kk + kh) * 64 + nn];
                    bf.y = Bs[buf][(kk + kh + 1) * 64 + nn];
                }
                acc[t] = __builtin_amdgcn_wmma_f32_16x16x4_f32(
                    false, a, false, bf, (short)0, acc[t], false, false);
            }
        }
        cp_wait();
        __syncthreads();
        buf ^= 1;
    }

    #pragma unroll
    for (int t = 0; t < 4; ++t) {
        const int n = n0 + t * 16 + ln;
        const float bv = bias[n];
        #pragma unroll
        for (int i = 0; i < 8; ++i) {
            const int m = m0 + i + ((lane < 16) ? 0 : 8);
            float v = acc[t][i] + bv;
            if (ACT == 1) v = GX[(size_t)m * N + n] * sigf(v);
            Y[(size_t)m * N + n] = v;
        }
    }
}

// ---------------------------------------------------------------------------
// Fused additive-attention: score=tanh(cd+qd)·Ws, masked softmax over LQ,
// att = sim @ q_repr; writes rnn_in row = [context_repr | att].
// One block (256 thr) per (b,p). bs cancels in softmax and is dropped.
// ---------------------------------------------------------------------------
__global__ __launch_bounds__(256) void attn_kernel(
    const float* __restrict__ cd, const float* __restrict__ qd,
    const float* __restrict__ qrep, const float* __restrict__ crep,
    const float* __restrict__ Ws, const float* __restrict__ qmask,
    float* __restrict__ rnn_in)
{
    __shared__ float cds[H_];
    __shared__ float wss[H_];
    __shared__ float sc[LQ_];
    __shared__ float sim[LQ_];

    const int tid  = threadIdx.x;
    const int lane = tid & 31;
    const int wave = tid >> 5;
    const int b = blockIdx.x >> 9;       // / LC_
    const int p = blockIdx.x & (LC_ - 1);
    const size_t crow = (size_t)(b * LC_ + p) * D_;

    cds[tid] = cd[crow + tid];
    wss[tid] = Ws[tid];
    __syncthreads();

    const float* qdb = qd + (size_t)b * LQ_ * H_;
    for (int qi = wave; qi < LQ_; qi += 8) {
        const float* qrow = qdb + qi * H_;
        float s = 0.f;
        #pragma unroll
        for (int hh = 0; hh < 8; ++hh) {
            const int h = lane + hh * 32;
            s += tanhf(cds[h] + qrow[h]) * wss[h];
        }
        #pragma unroll
        for (int off = 16; off > 0; off >>= 1) s += __shfl_xor(s, off, 32);
        if (lane == 0) sc[qi] = s;
    }
    __syncthreads();

    if (wave == 0) {
        const float* qm = qmask + b * LQ_;
        float s0 = (qm[lane] > 0.f)      ? sc[lane]      : -1e30f;
        float s1 = (qm[lane + 32] > 0.f) ? sc[lane + 32] : -1e30f;
        float mx = fmaxf(s0, s1);
        #pragma unroll
        for (int off = 16; off > 0; off >>= 1) mx = fmaxf(mx, __shfl_xor(mx, off, 32));
        float e0 = __expf(s0 - mx), e1 = __expf(s1 - mx);
        float sum = e0 + e1;
        #pragma unroll
        for (int off = 16; off > 0; off >>= 1) sum += __shfl_xor(sum, off, 32);
        const float inv = 1.f / sum;
        sim[lane] = e0 * inv;
        sim[lane + 32] = e1 * inv;
    }
    __syncthreads();

    const float* qr = qrep + (size_t)b * LQ_ * D_;
    float att = 0.f;
    #pragma unroll 8
    for (int q = 0; q < LQ_; ++q) att += sim[q] * qr[q * D_ + tid];

    float* rrow = rnn_in + (size_t)(b * LC_ + p) * (2 * D_);
    rrow[tid]      = crep[crow + tid];
    rrow[D_ + tid] = att;
}

// ---------------------------------------------------------------------------
// GRU scan: single workgroup (B=16 == one WMMA M-tile), 512 thr = 16 waves.
// Per step: hp = h@Whh^T + bhh via WMMA (each wave owns 48 of 768 columns,
// h staged in LDS), barrier, elementwise gates, barrier.
// ---------------------------------------------------------------------------
__global__ __launch_bounds__(512) void gru_kernel(
    const float* __restrict__ xproj, const float* __restrict__ Whh,
    const float* __restrict__ bhh, const int* __restrict__ ctx_len,
    float* __restrict__ out)
{
    __shared__ float hs[B_ * H_];        // 16 KB
    __shared__ float hp[B_ * 3 * H_];    // 48 KB
    __shared__ int   clen[B_];

    const int tid  = threadIdx.x;
    const int lane = tid & 31;
    const int wave = tid >> 5;           // 0..15
    const int ln   = lane & 15;
    const int kh   = (lane < 16) ? 0 : 2;

    for (int i = tid; i < B_ * H_; i += 512) hs[i] = 0.f;
    if (tid < B_) clen[tid] = ctx_len[tid];

    // Per-wave constants: 3 output tiles of 16 columns each.
    int ncol[3]; float bv[3]; const float* wrow[3];
    #pragma unroll
    for (int t = 0; t < 3; ++t) {
        ncol[t] = wave * 48 + t * 16 + ln;
        bv[t]   = bhh[ncol[t]];
        wrow[t] = Whh + (size_t)ncol[t] * H_;    // row of Whh (B^T access)
    }
    __syncthreads();

    for (int t = 0; t < LC_; ++t) {
        // hp = h @ Whh^T + bhh
        #pragma unroll
        for (int tau = 0; tau < 3; ++tau) {
            v8f acc = {};
            const float* wr = wrow[tau];
            for (int k = 0; k < H_; k += 4) {
                v2f a, bf;
                a.x  = hs[ln * H_ + k + kh];
                a.y  = hs[ln * H_ + k + kh + 1];
                bf.x = wr[k + kh];
                bf.y = wr[k + kh + 1];
                acc = __builtin_amdgcn_wmma_f32_16x16x4_f32(
                    false, a, false, bf, (short)0, acc, false, false);
            }
            #pragma unroll
            for (int i = 0; i < 8; ++i) {
                const int m = i + ((lane < 16) ? 0 : 8);
                hp[m * (3 * H_) + ncol[tau]] = acc[i] + bv[tau];
            }
        }
        __syncthreads();

        // Elementwise gates over 16x256 elements.
        for (int e = tid; e < B_ * H_; e += 512) {
            const int bb = e >> 8;
            const int j  = e & (H_ - 1);
            const float* xr = xproj + ((size_t)bb * LC_ + t) * (3 * H_);
            const float* hr = hp + bb * (3 * H_);
            const float r = sigf(xr[j] + hr[j]);
            const float z = sigf(xr[H_ + j] + hr[H_ + j]);
            const float n = tanhf(xr[2 * H_ + j] + r * hr[2 * H_ + j]);
            const float hn = (1.f - z) * n + z * hs[e];
            hs[e] = hn;
            out[((size_t)bb * LC_ + t) * H_ + j] = (t < clen[bb]) ? hn : 0.f;
        }
        __syncthreads();
    }
}

// ---------------------------------------------------------------------------
extern "C" void kernel_launch(void* const* d_in, const int* in_sizes, int n_in,
                              void* d_out, int out_size, void* d_ws, size_t ws_size,
                              hipStream_t stream)
{
    const float* ctx   = (const float*)d_in[0];
    const float* qrep  = (const float*)d_in[1];
    const int*   clen  = (const int*)d_in[2];
    const float* qmask = (const float*)d_in[3];
    const float* Wc    = (const float*)d_in[4];
    const float* bc    = (const float*)d_in[5];
    const float* Wq    = (const float*)d_in[6];
    const float* bq    = (const float*)d_in[7];
    const float* Ws    = (const float*)d_in[8];
    // d_in[9] = bs: constant shift, cancels in softmax.
    const float* Wg    = (const float*)d_in[10];
    const float* bg    = (const float*)d_in[11];
    const float* Wih   = (const float*)d_in[12];
    const float* Whh   = (const float*)d_in[13];
    const float* bih   = (const float*)d_in[14];
    const float* bhh   = (const float*)d_in[15];
    float* out = (float*)d_out;

    float* ws    = (float*)d_ws;
    float* cd    = ws;                                    // 8192*256
    float* qd    = cd + (size_t)B_ * LC_ * H_;            // 1024*256
    float* rnn   = qd + (size_t)B_ * LQ_ * H_;            // 8192*512
    float* gated = rnn + (size_t)B_ * LC_ * 2 * D_;       // 8192*512
    float* xp    = gated + (size_t)B_ * LC_ * 2 * D_;     // 8192*768

    const dim3 blk(256);
    // cd = ctx @ Wc + bc    : M=8192, N=256, K=256
    gemm_kernel<0, 0><<<dim3(4, 64), blk, 0, stream>>>(ctx, Wc, bc, nullptr, cd, 256, 256);
    // qd = q @ Wq + bq      : M=1024, N=256, K=256
    gemm_kernel<0, 0><<<dim3(4, 8), blk, 0, stream>>>(qrep, Wq, bq, nullptr, qd, 256, 256);
    // fused scores/softmax/att -> rnn_in = [ctx | att]
    attn_kernel<<<dim3(B_ * LC_), blk, 0, stream>>>(cd, qd, qrep, ctx, Ws, qmask, rnn);
    // gated = rnn * sigmoid(rnn @ Wg + bg) : M=8192, N=512, K=512
    gemm_kernel<1, 0><<<dim3(8, 64), blk, 0, stream>>>(rnn, Wg, bg, rnn, gated, 512, 512);
    // x_proj = gated @ Wih^T + bih : M=8192, N=768, K=512
    gemm_kernel<0, 1><<<dim3(12, 64), blk, 0, stream>>>(gated, Wih, bih, nullptr, xp, 768, 512);
    // GRU scan + output mask
    gru_kernel<<<dim3(1), dim3(512), 0, stream>>>(xp, Whh, bhh, clen, out);
}